// M2GNN_word_266287972522
// MI455X (gfx1250) — compile-verified
//
#include <hip/hip_runtime.h>
#include <math.h>

typedef __attribute__((ext_vector_type(2))) float v2f;
typedef __attribute__((ext_vector_type(8))) float v8f;

#define NEGV (-65535.0f)

__device__ __forceinline__ v8f wmma4(v2f a, v2f b, v8f c) {
  // V_WMMA_F32_16X16X4_F32 : D(16x16 f32) = A(16x4 f32) x B(4x16 f32) + C
  return __builtin_amdgcn_wmma_f32_16x16x4_f32(false, a, false, b, (short)0, c, false, false);
}

// Async global->LDS copy of 16 bytes per lane (GLOBAL_LOAD_ASYNC_TO_LDS_B128,
// tracked by ASYNCcnt).  lds_dst: generic pointer to __shared__ (low 32 bits
// are the LDS byte offset); gsrc: global address.
__device__ __forceinline__ void async_copy16(float4* lds_dst, const float4* gsrc) {
  unsigned loff = (unsigned)(unsigned long long)lds_dst;
  asm volatile("global_load_async_to_lds_b128 %0, %1, off"
               :: "v"(loff), "v"(gsrc)
               : "memory");
}

__device__ __forceinline__ void wait_async0() {
#if __has_builtin(__builtin_amdgcn_s_wait_asynccnt)
  __builtin_amdgcn_s_wait_asynccnt(0);
#else
  asm volatile("s_wait_asynccnt 0" ::: "memory");
#endif
}

// ---------------------------------------------------------------------------
// Copy input B_matrix (1x8x128) into workspace (inputs must not be mutated)
__global__ void k_init_B(const float* __restrict__ Bin, float* __restrict__ Bws) {
  for (int i = threadIdx.x; i < 1024; i += 256) Bws[i] = Bin[i];
}

// ---------------------------------------------------------------------------
// low_new[r, :] = embed[idx[r], :] @ S   for r in [0, R),  R = n_user * n
// One wave computes a 16-row x 64-col tile: 4 N-tiles x 16 K-steps of
// v_wmma_f32_16x16x4_f32.  S (64x64, 16KB) staged in LDS per block.
__global__ void k_gather_gemm(const float* __restrict__ embed, const int* __restrict__ idx,
                              const float* __restrict__ S, float* __restrict__ low,
                              int n, int R) {
  __shared__ __align__(16) float sS[64 * 64];
  for (int i = threadIdx.x; i < 1024; i += blockDim.x)
    ((float4*)sS)[i] = ((const float4*)S)[i];
  __syncthreads();

  int wave = threadIdx.x >> 5, lane = threadIdx.x & 31;
  int m = lane & 15, kh = lane >> 4;        // A: lanes 0-15 K=0,1 ; lanes 16-31 K=2,3
  int tile = blockIdx.x * 8 + wave;
  int row0 = tile * 16;
  if (row0 >= R) return;

  int r = row0 + m;
  if (r >= R) r = R - 1;
  int b = r / n, j = r - b * n;
  const float* src = embed + (size_t)idx[(size_t)b * n + j] * 64;

  v2f A[16];
#pragma unroll
  for (int kk = 0; kk < 16; ++kk) {
    int k0 = kk * 4 + kh * 2;
    A[kk].x = src[k0];
    A[kk].y = src[k0 + 1];
  }

  float* dst = low + (size_t)row0 * 64;
#pragma unroll
  for (int t = 0; t < 4; ++t) {
    v8f acc = {};
#pragma unroll
    for (int kk = 0; kk < 16; ++kk) {
      int k0 = kk * 4 + kh * 2;
      v2f bf;
      bf.x = sS[k0 * 64 + t * 16 + m];        // B: row K striped across lanes (N)
      bf.y = sS[(k0 + 1) * 64 + t * 16 + m];
      acc = wmma4(A[kk], bf, acc);
    }
#pragma unroll
    for (int v = 0; v < 8; ++v) {             // D: vgpr v = row v (+8 upper lanes)
      int mm = v + 8 * kh;
      if (row0 + mm < R) dst[(size_t)mm * 64 + t * 16 + m] = acc[v];
    }
  }
}

// ---------------------------------------------------------------------------
// W = softmax over MAX_LEN (mask j>=n -> NEG), stored zero-padded as 16x128.
// Also zeroes the delta accumulator.  Single block.
__global__ void k_softmax_W(const float* __restrict__ Bws, float* __restrict__ Wp,
                            float* __restrict__ delta, int n, int Kc) {
  int tid = threadIdx.x;
  for (int i = tid; i < 2048; i += 256) { Wp[i] = 0.f; delta[i] = 0.f; }
  __syncthreads();
  int k = tid >> 5, lane = tid & 31;        // one wave per capsule row
  if (k < Kc) {
    float vals[4];
    float mx = -INFINITY;
#pragma unroll
    for (int q = 0; q < 4; ++q) {
      int j = lane + 32 * q;
      float x = (j < n) ? Bws[k * 128 + j] : NEGV;
      vals[q] = x;
      mx = fmaxf(mx, x);
    }
    for (int off = 16; off > 0; off >>= 1) mx = fmaxf(mx, __shfl_xor(mx, off, 32));
    float s = 0.f;
#pragma unroll
    for (int q = 0; q < 4; ++q) { vals[q] = expf(vals[q] - mx); s += vals[q]; }
    for (int off = 16; off > 0; off >>= 1) s += __shfl_xor(s, off, 32);
    float inv = 1.f / s;
#pragma unroll
    for (int q = 0; q < 4; ++q) {
      int j = lane + 32 * q;
      if (j < n) Wp[k * 128 + j] = vals[q] * inv;
    }
  }
}

// ---------------------------------------------------------------------------
// One routing iteration.  Block = 128 threads (4 waves).  Double-buffered LDS:
// batch b+gridDim.x's low_new tile is prefetched with GLOBAL_LOAD_ASYNC_TO_LDS_B128
// (ASYNCcnt) while WMMAs run on batch b:
//   high = W @ low_new (each wave owns one 16-col tile, 16 WMMA),
//   squash rows, optionally store h, and
//   delta-tile accumulated in WMMA C registers across all batches of the block;
//   a single global atomicAdd pass per block at the end.
__global__ void k_routing(const float* __restrict__ low, const float* __restrict__ Wp,
                          float* __restrict__ deltag, float* __restrict__ allu,
                          int n, int B, int storeh, int kofs) {
  __shared__ __align__(16) float slow[2][64 * 64];  // 2 x 16 KB
  __shared__ __align__(16) float shigh[16 * 64];    // 4 KB
  __shared__ float ssc[8];

  int tid = threadIdx.x;
  int wave = tid >> 5, lane = tid & 31;
  int m = lane & 15, kh = lane >> 4;

  v2f AW[16];                       // W fragments: identical for every batch
#pragma unroll
  for (int kk = 0; kk < 16; ++kk) {
    int k0 = kk * 4 + kh * 2;
    AW[kk].x = Wp[m * 128 + k0];
    AW[kk].y = Wp[m * 128 + k0 + 1];
  }

  v8f dacc = {};                    // this wave's delta tile, summed over batches
  int nv = n * 16;                  // float4 count of valid rows
  float4 z4; z4.x = z4.y = z4.z = z4.w = 0.f;

  // Zero the batch-invariant padding rows (>= n) of both buffers once.
  for (int i = nv + tid; i < 1024; i += 128) {
    ((float4*)slow[0])[i] = z4;
    ((float4*)slow[1])[i] = z4;
  }

  int b = blockIdx.x;
  int buf = 0;
  if (b < B) {                      // prefetch first tile
    const float4* g = (const float4*)(low + (size_t)b * n * 64);
    float4* d = (float4*)slow[0];
    for (int i = tid; i < nv; i += 128) async_copy16(d + i, g + i);
  }

  for (; b < B; b += gridDim.x) {
    wait_async0();                  // this wave's async writes landed in LDS
    __syncthreads();                // publish tile block-wide; also guarantees
                                    // nobody still reads the buffer we prefetch next
    int bn = b + gridDim.x;
    if (bn < B) {                   // prefetch next batch into the other buffer
      const float4* g = (const float4*)(low + (size_t)bn * n * 64);
      float4* d = (float4*)slow[buf ^ 1];
      for (int i = tid; i < nv; i += 128) async_copy16(d + i, g + i);
    }

    const float* sl = slow[buf];

    // high = W @ low_new ; wave owns cols [wave*16, wave*16+16)
    v8f acc = {};
#pragma unroll
    for (int kk = 0; kk < 16; ++kk) {
      int k0 = kk * 4 + kh * 2;
      v2f bf;
      bf.x = sl[k0 * 64 + wave * 16 + m];
      bf.y = sl[(k0 + 1) * 64 + wave * 16 + m];
      acc = wmma4(AW[kk], bf, acc);
    }
#pragma unroll
    for (int v = 0; v < 8; ++v)
      shigh[(v + 8 * kh) * 64 + wave * 16 + m] = acc[v];     // rows >= Kc are 0
    __syncthreads();

    // squash rows 0..7 : x *= sq/(1+sq)/sqrt(sq+1e-9)
    if (tid < 8) {
      float sq = 0.f;
      const float* hr = shigh + tid * 64;
      for (int dd = 0; dd < 64; ++dd) sq += hr[dd] * hr[dd];
      ssc[tid] = sq / ((1.f + sq) * sqrtf(sq + 1e-9f));
    }
    __syncthreads();
    for (int i = tid; i < 512; i += 128) shigh[i] *= ssc[i >> 6];
    __syncthreads();

    if (storeh) {                   // last iteration: emit h (rows Kc..7 already 0)
      float* hp = allu + ((size_t)b * 24 + kofs) * 64;
      for (int i = tid; i < 512; i += 128) hp[i] = shigh[i];
    }

    // delta[k, j] += sum_d high[k,d] * low_new[j,d] ; wave owns j-tile = wave
#pragma unroll
    for (int kk = 0; kk < 16; ++kk) {
      int k0 = kk * 4 + kh * 2;
      v2f a, bf;
      a.x = shigh[m * 64 + k0];
      a.y = shigh[m * 64 + k0 + 1];
      bf.x = sl[(wave * 16 + m) * 64 + k0];
      bf.y = sl[(wave * 16 + m) * 64 + k0 + 1];
      dacc = wmma4(a, bf, dacc);
    }
    buf ^= 1;
  }

#pragma unroll
  for (int v = 0; v < 8; ++v)
    atomicAdd(&deltag[(v + 8 * kh) * 128 + wave * 16 + m], dacc[v]);
}

// ---------------------------------------------------------------------------
__global__ void k_apply_delta(float* __restrict__ Bws, const float* __restrict__ delta,
                              int n, int Kc) {
  for (int i = threadIdx.x; i < 1024; i += 256) {
    int k = i >> 7, j = i & 127;
    if (k < Kc && j < n) Bws[i] += delta[k * 128 + j];
  }
}

// ---------------------------------------------------------------------------
// Attention: s_k = (tanh(u_k.M1).M2)^2, att = softmax_k, out = sum att_k u_k.
// One wave per batch; u (24x64, padded to 32x64) staged in LDS; u.M1 via WMMA.
__global__ void k_attention(const float* __restrict__ allu, const float* __restrict__ M1g,
                            const float* __restrict__ M2g, float* __restrict__ out, int B) {
  __shared__ __align__(16) float sM1[64 * 64];
  __shared__ float sM2[64];
  __shared__ __align__(16) float su[4][32 * 64];
  __shared__ float ss[4][32];

  for (int i = threadIdx.x; i < 1024; i += blockDim.x)
    ((float4*)sM1)[i] = ((const float4*)M1g)[i];
  if (threadIdx.x < 64) sM2[threadIdx.x] = M2g[threadIdx.x];
  __syncthreads();

  int wave = threadIdx.x >> 5, lane = threadIdx.x & 31;
  int m = lane & 15, kh = lane >> 4;
  int b = blockIdx.x * 4 + wave;
  if (b >= B) return;

  float* u = su[wave];
  float4* u4 = (float4*)u;
  const float4* g = (const float4*)(allu + (size_t)b * 24 * 64);
  for (int i = lane; i < 384; i += 32) u4[i] = g[i];
  float4 z4; z4.x = z4.y = z4.z = z4.w = 0.f;
  for (int i = 384 + lane; i < 512; i += 32) u4[i] = z4;     // pad rows 24..31
  __builtin_amdgcn_wave_barrier();

#pragma unroll
  for (int mt = 0; mt < 2; ++mt) {
    float p[8];
#pragma unroll
    for (int v = 0; v < 8; ++v) p[v] = 0.f;
#pragma unroll
    for (int t = 0; t < 4; ++t) {
      v8f acc = {};
#pragma unroll
      for (int kk = 0; kk < 16; ++kk) {
        int k0 = kk * 4 + kh * 2;
        v2f a, bf;
        a.x = u[(mt * 16 + m) * 64 + k0];
        a.y = u[(mt * 16 + m) * 64 + k0 + 1];
        bf.x = sM1[k0 * 64 + t * 16 + m];
        bf.y = sM1[(k0 + 1) * 64 + t * 16 + m];
        acc = wmma4(a, bf, acc);
      }
      float m2v = sM2[t * 16 + m];
#pragma unroll
      for (int v = 0; v < 8; ++v) p[v] += tanhf(acc[v]) * m2v;
    }
#pragma unroll
    for (int v = 0; v < 8; ++v) {              // reduce over 16 lanes of each half
      float x = p[v];
      for (int off = 8; off > 0; off >>= 1) x += __shfl_xor(x, off, 16);
      if (m == 0) ss[wave][mt * 16 + v + 8 * kh] = x;
    }
  }
  __builtin_amdgcn_wave_barrier();

  float sv = -INFINITY;
  if (lane < 24) { float t = ss[wave][lane]; sv = t * t; }   // GAMMA = 2
  float mx = sv;
  for (int off = 16; off > 0; off >>= 1) mx = fmaxf(mx, __shfl_xor(mx, off, 32));
  float e = (lane < 24) ? expf(sv - mx) : 0.f;
  float sum = e;
  for (int off = 16; off > 0; off >>= 1) sum += __shfl_xor(sum, off, 32);
  float att = e / sum;
  if (lane < 24) ss[wave][lane] = att;
  __builtin_amdgcn_wave_barrier();

  float o0 = 0.f, o1 = 0.f;
  for (int k = 0; k < 24; ++k) {
    float a = ss[wave][k];
    o0 += a * u[k * 64 + lane];
    o1 += a * u[k * 64 + 32 + lane];
  }
  out[(size_t)b * 64 + lane] = o0;
  out[(size_t)b * 64 + 32 + lane] = o1;
}

// ---------------------------------------------------------------------------
__global__ void k_review_mean(const float* __restrict__ tag, const int* __restrict__ idx,
                              float* __restrict__ out, int Nrev, int L) {
  int i = blockIdx.x * 4 + (threadIdx.x >> 6);
  int dd = threadIdx.x & 63;
  if (i >= Nrev) return;
  const int* row = idx + (size_t)i * L;
  float s = 0.f;
  for (int t = 0; t < L; ++t) s += tag[(size_t)row[t] * 64 + dd];
  out[(size_t)i * 64 + dd] = s / (float)L;
}

// ---------------------------------------------------------------------------
extern "C" void kernel_launch(void* const* d_in, const int* in_sizes, int n_in,
                              void* d_out, int out_size, void* d_ws, size_t ws_size,
                              hipStream_t stream) {
  const float* review = (const float*)d_in[0];
  const float* tag    = (const float*)d_in[1];
  const int* idx_urt  = (const int*)d_in[2];
  const int* idx_uqt  = (const int*)d_in[3];
  const int* idx_uprt = (const int*)d_in[4];
  const int* idx_rht  = (const int*)d_in[5];
  const float* Bin    = (const float*)d_in[6];
  const float* S      = (const float*)d_in[7];
  const float* M1     = (const float*)d_in[8];
  const float* M2     = (const float*)d_in[9];

  int n_user = in_sizes[2] / 64;               // idx_urt is (n_user, 64)
  int n1 = 64;
  int n2 = in_sizes[3] / n_user;               // 32
  int n3 = in_sizes[4] / n_user;               // 50
  const int Lr = 20;
  int n_rev = in_sizes[5] / Lr;

  float* w = (float*)d_ws;
  float* wsB = w;                              // 1024
  float* wsW = w + 1024;                       // 16x128
  float* wsD = w + 3072;                       // 16x128
  float* wsLow = w + 5120;                     // n_user*64*64
  float* wsU = wsLow + (size_t)n_user * 64 * 64;  // n_user*24*64
  size_t need = (5120 + (size_t)n_user * 64 * 64 + (size_t)n_user * 24 * 64) * sizeof(float);
  if (ws_size < need) return;

  float* out_user = (float*)d_out;
  float* out_rev = out_user + (size_t)n_user * 64;

  k_init_B<<<1, 256, 0, stream>>>(Bin, wsB);

  const float* embeds[3] = {review, tag, review};
  const int* idxs[3] = {idx_urt, idx_uqt, idx_uprt};
  int ns[3] = {n1, n2, n3};

  for (int r = 0; r < 3; ++r) {
    int n = ns[r];
    int kc = 0;
    for (int t = n; t > 1; t >>= 1) kc++;      // floor(log2(n))
    if (kc < 1) kc = 1;
    if (kc > 8) kc = 8;
    long R = (long)n_user * n;
    int gblocks = (int)((R + 16 * 8 - 1) / (16 * 8));
    k_gather_gemm<<<gblocks, 256, 0, stream>>>(embeds[r], idxs[r], S, wsLow, n, (int)R);
    for (int it = 0; it < 3; ++it) {
      k_softmax_W<<<1, 256, 0, stream>>>(wsB, wsW, wsD, n, kc);
      k_routing<<<625, 128, 0, stream>>>(wsLow, wsW, wsD, wsU, n, n_user,
                                         (it == 2) ? 1 : 0, r * 8);
      k_apply_delta<<<1, 256, 0, stream>>>(wsB, wsD, n, kc);
    }
  }

  k_attention<<<(n_user + 3) / 4, 256, 0, stream>>>(wsU, M1, M2, out_user, n_user);
  k_review_mean<<<(n_rev + 3) / 4, 256, 0, stream>>>(tag, idx_rht, out_rev, n_rev, Lr);
}